// ActorNetwork_71442486001991
// MI455X (gfx1250) — compile-verified
//
#include <hip/hip_runtime.h>
#include <hip/hip_bf16.h>

typedef __attribute__((ext_vector_type(2))) float v2f;
typedef __attribute__((ext_vector_type(8))) float v8f;

static inline long long cdivll(long long a, long long b) { return (a + b - 1) / b; }

// ---------------------------------------------------------------------------
// Conv3d (1,3,1 kernel, VALID) + bias + ReLU.
// x: (4, 64, 512, 32), w: (32,4,1,3,1), b: (32) -> t: (32, 64, 510, 32)
// ---------------------------------------------------------------------------
__global__ void conv_relu_kernel(const float* __restrict__ x,
                                 const float* __restrict__ w,
                                 const float* __restrict__ b,
                                 float* __restrict__ t, int total)
{
    int idx = blockIdx.x * blockDim.x + threadIdx.x;
    if (idx >= total) return;
    int ww = idx & 31;
    int r  = idx >> 5;
    int h  = r % 510; r /= 510;
    int dd = r & 63;
    int o  = r >> 6;
    float acc = b[o];
#pragma unroll
    for (int i = 0; i < 4; ++i) {
#pragma unroll
        for (int kh = 0; kh < 3; ++kh) {
            float xv = x[((i * 64 + dd) * 512 + (h + kh)) * 32 + ww];
            acc = fmaf(xv, w[(o * 4 + i) * 3 + kh], acc);
        }
    }
    t[idx] = fmaxf(acc, 0.0f);
}

// ---------------------------------------------------------------------------
// WMMA fp32 gram kernel: G(dpad x dpad) += A A^T over a K-chunk, where
// A = unfold(tensor, mode). The wave-uniform k counter is decomposed once per
// chunk and advanced with carry increments (no division in the loop).
// K and kChunk must be multiples of 8. One wave per 16x16 tile per chunk;
// two V_WMMA_F32_16X16X4_F32 per iteration.
//   PAD : d is not a multiple of 16 -> zero out padded rows (flag multiply)
//   VEC2: innermost remaining stride==1 and extent even -> contiguous aligned
//         K-pairs, load each operand half as one b64.
// ---------------------------------------------------------------------------
template <bool PAD, bool VEC2>
__global__ void gram_wmma_kernel(const float* __restrict__ T, float* __restrict__ G,
                                 int d, int dpad, int tilesJ,
                                 int rowStride, int o1, int o2,
                                 int s0, int s1, int s2,
                                 int K, int kChunk)
{
    const int lane = threadIdx.x;                 // 32 threads = 1 wave
    const int tj   = blockIdx.x % tilesJ;
    const int ti   = blockIdx.x / tilesJ;
    const int rA   = ti * 16 + (lane & 15);       // A row (M = lane&15)
    const int rB   = tj * 16 + (lane & 15);       // B column (N = lane&15)
    const bool hi  = (lane & 16) != 0;            // K-half select

    float fA = 1.0f, fB = 1.0f;
    int baseA, baseB;
    if (PAD) {
        fA = (rA < d) ? 1.0f : 0.0f;
        fB = (rB < d) ? 1.0f : 0.0f;
        baseA = ((rA < d) ? rA : 0) * rowStride;
        baseB = ((rB < d) ? rB : 0) * rowStride;
    } else {
        baseA = rA * rowStride;
        baseB = rB * rowStride;
    }

    int kBeg = blockIdx.y * kChunk;
    int kEnd = kBeg + kChunk;
    if (kEnd > K) kEnd = K;

    // one-time decomposition of the (wave-uniform) chunk start
    int q2  = kBeg % o2;
    int rem = kBeg / o2;
    int q1  = rem % o1;
    int lin = (rem / o1) * s0 + q1 * s1 + q2 * s2;
    const int c2 = s1 - o2 * s2;                  // add when q2 wraps
    const int c1 = s0 - o1 * s1;                  // add when q1 wraps

    v8f acc = {0.f, 0.f, 0.f, 0.f, 0.f, 0.f, 0.f, 0.f};

    for (int k = kBeg; k < kEnd; k += 8) {
        int off[8];
#pragma unroll
        for (int j = 0; j < 8; ++j) {             // 8 carry-increments, no division
            off[j] = lin;
            ++q2; lin += s2;
            if (q2 == o2) {
                q2 = 0; lin += c2;
                ++q1;
                if (q1 == o1) { q1 = 0; lin += c1; }
            }
        }
#pragma unroll
        for (int half = 0; half < 2; ++half) {
            const int ka = hi ? off[4 * half + 2] : off[4 * half];
            v2f av, bv;
            if (VEC2) {
                av = *(const v2f*)(T + baseA + ka);    // one b64 per operand
                bv = *(const v2f*)(T + baseB + ka);
            } else {
                const int kb = hi ? off[4 * half + 3] : off[4 * half + 1];
                av.x = T[baseA + ka]; av.y = T[baseA + kb];
                bv.x = T[baseB + ka]; bv.y = T[baseB + kb];
            }
            if (PAD) { av *= fA; bv *= fB; }
            acc = __builtin_amdgcn_wmma_f32_16x16x4_f32(false, av, false, bv,
                                                        (short)0, acc, false, false);
        }
        __builtin_prefetch(&T[baseA + lin], 0, 3);   // global_prefetch_b8 (near)
    }

    // C/D layout: VGPR v -> row = v + 8*(lane>>4); col = lane&15
    const int col   = tj * 16 + (lane & 15);
    const int rbase = ti * 16 + (hi ? 8 : 0);
#pragma unroll
    for (int v = 0; v < 8; ++v)
        atomicAdd(&G[(rbase + v) * dpad + col], acc[v]);
}

// ---------------------------------------------------------------------------
// mode_dot: out = tensor x_mode F^T  (F is dm x r row-major; M[j,i]=F[i*r+j]).
// Input contiguous (d0,d1,d2,d3); output contiguous with dims[mode] -> r.
// ---------------------------------------------------------------------------
__global__ void mode_dot_kernel(const float* __restrict__ in, const float* __restrict__ F,
                                float* __restrict__ out,
                                int d0, int d1, int d2, int d3,
                                int mode, int r, int total)
{
    int idx = blockIdx.x * blockDim.x + threadIdx.x;
    if (idx >= total) return;
    int od1 = (mode == 1) ? r : d1;
    int od2 = (mode == 2) ? r : d2;
    int od3 = (mode == 3) ? r : d3;
    int tmp = idx;
    int a3 = tmp % od3; tmp /= od3;
    int a2 = tmp % od2; tmp /= od2;
    int a1 = tmp % od1; tmp /= od1;
    int a0 = tmp;

    int s3 = 1;
    int s2 = d3;
    int s1 = d2 * d3;
    int s0 = d1 * d2 * d3;

    int am, dm, smode;
    if (mode == 0)      { am = a0; smode = s0; dm = d0; a0 = 0; }
    else if (mode == 1) { am = a1; smode = s1; dm = d1; a1 = 0; }
    else if (mode == 2) { am = a2; smode = s2; dm = d2; a2 = 0; }
    else                { am = a3; smode = s3; dm = d3; a3 = 0; }

    int base = a0 * s0 + a1 * s1 + a2 * s2 + a3 * s3;
    float acc = 0.0f;
    for (int i = 0; i < dm; ++i)
        acc = fmaf(in[base + i * smode], F[i * r + am], acc);
    out[idx] = acc;
}

// ---------------------------------------------------------------------------
// Top-r eigenvectors of symmetric PSD S (n x n, row stride = stride) by
// orthogonal (subspace) iteration with modified Gram-Schmidt. Single WG.
// Output F: (n x r) row-major, descending eigenvalue order, sign-normalized.
// ---------------------------------------------------------------------------
__global__ __launch_bounds__(512)
void subspace_eig_kernel(const float* __restrict__ S, int n, int stride, int r,
                         float* __restrict__ F, int niter)
{
    __shared__ float Q[512 * 8];
    __shared__ float Y[512 * 8];
    __shared__ float red[512];
    __shared__ float sgn[8];

    const int tid  = threadIdx.x;
    const int nthr = blockDim.x;

    // deterministic pseudo-random init
    for (int i = tid; i < n; i += nthr) {
        for (int j = 0; j < r; ++j) {
            unsigned u = (unsigned)(i * r + j) * 2654435761u + 12345u;
            u ^= u >> 13; u *= 2246822519u; u ^= u >> 16;
            Q[i * r + j] = (float)(u & 0xFFFFu) * (1.0f / 65536.0f) - 0.5f;
        }
    }
    __syncthreads();

    for (int it = 0; it < niter; ++it) {
        // Y = S * Q : one matrix row per thread, all r columns in registers
        for (int i = tid; i < n; i += nthr) {
            float yv[8];
#pragma unroll
            for (int j = 0; j < 8; ++j) yv[j] = 0.0f;
            const float* Srow = S + i * stride;
            for (int k = 0; k < n; ++k) {
                float sv = Srow[k];
#pragma unroll 8
                for (int j = 0; j < r; ++j) yv[j] = fmaf(sv, Q[k * r + j], yv[j]);
            }
            for (int j = 0; j < r; ++j) Y[i * r + j] = yv[j];
        }
        __syncthreads();
        // MGS: Q = qr(Y)
        for (int j = 0; j < r; ++j) {
            float part = 0.0f;
            for (int i = tid; i < n; i += nthr) { float v = Y[i * r + j]; part = fmaf(v, v, part); }
            red[tid] = part; __syncthreads();
            for (int off = nthr >> 1; off > 0; off >>= 1) {
                if (tid < off) red[tid] += red[tid + off];
                __syncthreads();
            }
            float inv = rsqrtf(red[0] + 1e-30f);
            __syncthreads();
            for (int i = tid; i < n; i += nthr) Q[i * r + j] = Y[i * r + j] * inv;
            __syncthreads();
            for (int l = j + 1; l < r; ++l) {
                part = 0.0f;
                for (int i = tid; i < n; i += nthr) part = fmaf(Q[i * r + j], Y[i * r + l], part);
                red[tid] = part; __syncthreads();
                for (int off = nthr >> 1; off > 0; off >>= 1) {
                    if (tid < off) red[tid] += red[tid + off];
                    __syncthreads();
                }
                float dp = red[0]; __syncthreads();
                for (int i = tid; i < n; i += nthr) Y[i * r + l] -= dp * Q[i * r + j];
                __syncthreads();
            }
        }
    }

    if (tid < r) sgn[tid] = (Q[tid] < 0.0f) ? -1.0f : 1.0f;  // convention: F[0][j] >= 0
    __syncthreads();
    for (int i = tid; i < n; i += nthr)
        for (int j = 0; j < r; ++j)
            F[i * r + j] = Q[i * r + j] * sgn[j];
}

// ---------------------------------------------------------------------------
// FC (30 x 384 weights, core has 192 valid entries) + softmax (+ renorm).
// ---------------------------------------------------------------------------
__global__ void fc_softmax_kernel(const float* __restrict__ core,
                                  const float* __restrict__ fcw,
                                  const float* __restrict__ fcb,
                                  float* __restrict__ out)
{
    __shared__ float logits[32];
    __shared__ float red[32];
    int tid = threadIdx.x;
    float v = -3.4e38f;
    if (tid < 30) {
        float s = fcb[tid];
        for (int k = 0; k < 192; ++k) s = fmaf(fcw[tid * 384 + k], core[k], s);
        v = s;
    }
    logits[tid] = v;
    red[tid] = v;
    __syncthreads();
    for (int off = 16; off > 0; off >>= 1) {
        if (tid < off) red[tid] = fmaxf(red[tid], red[tid + off]);
        __syncthreads();
    }
    float mx = red[0]; __syncthreads();
    float e = (tid < 30) ? __expf(logits[tid] - mx) : 0.0f;
    red[tid] = e; __syncthreads();
    for (int off = 16; off > 0; off >>= 1) {
        if (tid < off) red[tid] += red[tid + off];
        __syncthreads();
    }
    float sum = red[0];
    if (tid < 30) out[tid] = e / sum;
}

// ---------------------------------------------------------------------------
// Host-side orchestration
// ---------------------------------------------------------------------------
static void launch_gram(const float* T, float* G, const int dims[4], int mode,
                        hipStream_t stream)
{
    int d = dims[mode];
    int dpad = ((d + 15) / 16) * 16;
    int str[4];
    str[3] = 1;
    str[2] = dims[3];
    str[1] = dims[2] * dims[3];
    str[0] = dims[1] * dims[2] * dims[3];
    int rowStride = str[mode];
    int os[3], ss[3], c = 0;
    for (int m = 0; m < 4; ++m)
        if (m != mode) { os[c] = dims[m]; ss[c] = str[m]; ++c; }
    int K = os[0] * os[1] * os[2];          // always a multiple of 8 here
    hipMemsetAsync(G, 0, (size_t)dpad * dpad * sizeof(float), stream);
    const int kChunk = 4096;                // multiple of 8
    int tiles = dpad / 16;
    dim3 grid((unsigned)(tiles * tiles), (unsigned)cdivll(K, kChunk));

    bool pad  = (d & 15) != 0;
    bool vec2 = (ss[2] == 1) && ((os[2] & 1) == 0);  // contiguous, even inner extent
    if (pad) {
        if (vec2) gram_wmma_kernel<true, true><<<grid, 32, 0, stream>>>(
            T, G, d, dpad, tiles, rowStride, os[1], os[2], ss[0], ss[1], ss[2], K, kChunk);
        else      gram_wmma_kernel<true, false><<<grid, 32, 0, stream>>>(
            T, G, d, dpad, tiles, rowStride, os[1], os[2], ss[0], ss[1], ss[2], K, kChunk);
    } else {
        if (vec2) gram_wmma_kernel<false, true><<<grid, 32, 0, stream>>>(
            T, G, d, dpad, tiles, rowStride, os[1], os[2], ss[0], ss[1], ss[2], K, kChunk);
        else      gram_wmma_kernel<false, false><<<grid, 32, 0, stream>>>(
            T, G, d, dpad, tiles, rowStride, os[1], os[2], ss[0], ss[1], ss[2], K, kChunk);
    }
}

static void launch_mode_dot(const float* in, const int indims[4], int mode, int r,
                            const float* F, float* out, int outdims[4],
                            hipStream_t stream)
{
    for (int i = 0; i < 4; ++i) outdims[i] = (i == mode) ? r : indims[i];
    int total = outdims[0] * outdims[1] * outdims[2] * outdims[3];
    mode_dot_kernel<<<(unsigned)cdivll(total, 256), 256, 0, stream>>>(
        in, F, out, indims[0], indims[1], indims[2], indims[3], mode, r, total);
}

extern "C" void kernel_launch(void* const* d_in, const int* in_sizes, int n_in,
                              void* d_out, int out_size, void* d_ws, size_t ws_size,
                              hipStream_t stream)
{
    (void)in_sizes; (void)n_in; (void)out_size; (void)ws_size;
    const float* x      = (const float*)d_in[0];
    const float* conv_w = (const float*)d_in[1];
    const float* conv_b = (const float*)d_in[2];
    const float* fc_w   = (const float*)d_in[3];
    const float* fc_b   = (const float*)d_in[4];
    float* out = (float*)d_out;
    float* ws  = (float*)d_ws;

    const int D0 = 32, D1 = 64, D2 = 510, D3 = 32;
    const int TN = D0 * D1 * D2 * D3;        // 33,423,360

    size_t off = 0;
    float* T    = ws + off; off += (size_t)TN;
    float* B0   = ws + off; off += 1100800;  // >= 32*2*510*32 = 1,044,480
    float* B1   = ws + off; off += 1100800;
    float* G    = ws + off; off += 512 * 512;
    float* F0   = ws + off; off += 256;      // 32*8
    float* F1   = ws + off; off += 128;      // 64*2
    float* F2   = ws + off; off += 3072;     // 510*6
    float* F3   = ws + off; off += 64;       // 32*2
    float* CORE = ws + off; off += 256;      // 192

    // 1. conv + bias + relu -> T
    conv_relu_kernel<<<(unsigned)cdivll(TN, 256), 256, 0, stream>>>(
        x, conv_w, conv_b, T, TN);

    const int tdims[4] = {D0, D1, D2, D3};
    const int ranks[4] = {8, 2, 6, 2};
    float* Fs[4] = {F0, F1, F2, F3};
    const int EIG_ITERS = 32;

    // 2. init factors: top-r eigenvectors of unfold(T,n) * unfold(T,n)^T
    for (int n = 0; n < 4; ++n) {
        launch_gram(T, G, tdims, n, stream);
        int d = tdims[n], dpad = ((d + 15) / 16) * 16;
        subspace_eig_kernel<<<1, 512, 0, stream>>>(G, d, dpad, ranks[n], Fs[n], EIG_ITERS);
    }

    // 3. HOOI sweeps: project along all modes != n (largest reduction first),
    //    then top-r_n left singular basis via gram + subspace iteration.
    const int order[4][3] = { {2, 1, 3}, {2, 3, 0}, {1, 3, 0}, {2, 1, 0} };
    for (int it = 0; it < 10; ++it) {
        for (int n = 0; n < 4; ++n) {
            const float* cur = T;
            int cdims[4] = {D0, D1, D2, D3};
            float* bufs[2] = {B0, B1};
            int bi = 0;
            for (int q = 0; q < 3; ++q) {
                int m = order[n][q];
                int odims[4];
                launch_mode_dot(cur, cdims, m, ranks[m], Fs[m], bufs[bi], odims, stream);
                cur = bufs[bi];
                for (int i = 0; i < 4; ++i) cdims[i] = odims[i];
                bi ^= 1;
            }
            launch_gram(cur, G, cdims, n, stream);
            int d = cdims[n], dpad = ((d + 15) / 16) * 16;
            subspace_eig_kernel<<<1, 512, 0, stream>>>(G, d, dpad, ranks[n], Fs[n], EIG_ITERS);
        }
    }

    // 4. core = T x_0 F0^T x_1 F1^T x_2 F2^T x_3 F3^T (cheapest order)
    {
        const float* cur = T;
        int cdims[4] = {D0, D1, D2, D3};
        float* bufs[2] = {B0, B1};
        int bi = 0;
        const int corder[4] = {2, 1, 3, 0};
        for (int q = 0; q < 4; ++q) {
            int m = corder[q];
            int odims[4];
            float* dst = (q == 3) ? CORE : bufs[bi];
            launch_mode_dot(cur, cdims, m, ranks[m], Fs[m], dst, odims, stream);
            cur = dst;
            for (int i = 0; i < 4; ++i) cdims[i] = odims[i];
            bi ^= 1;
        }
    }

    // 5. FC + softmax (+ renorm, idempotent)
    fc_softmax_kernel<<<1, 32, 0, stream>>>(CORE, fc_w, fc_b, out);
}